// RoIPlusAvg_31327491457502
// MI455X (gfx1250) — compile-verified
//
#include <hip/hip_runtime.h>

// RoI-align (8x8 bilinear grid) + 2x2 average pool -> (N, 512, 7, 7) fp32.
// B=2, C=512, H=50, W=75, N=2000, spatial_scale = 1/16.
// Strategy:
//   K1: transpose features BCHW -> BHWC into d_ws (channel-contiguous gathers).
//   K2: one block per RoI. Stage 64 sample bases/weights in LDS; per 128-ch
//       chunk, blend bilinear samples into LDS V[64][128]; then the 2x2 avg
//       pool is the dense matmul O(49xC) = M(49x64) x V(64xC), executed as
//       chained V_WMMA_F32_16X16X4_F32 (K=64 -> 16 steps), 16 channels/wave.

static constexpr int B_ = 2, C_ = 512, H_ = 50, W_ = 75;
static constexpr int CHUNK = 128;      // channels per LDS staging chunk
static constexpr int VSTR  = CHUNK + 1; // padded LDS row stride (bank-safe)
static constexpr int MSTR  = 65;        // padded M row stride (bank-safe)

typedef __attribute__((ext_vector_type(2))) float v2f;
typedef __attribute__((ext_vector_type(8))) float v8f;

__global__ void transpose_bchw_to_bhwc(const float* __restrict__ in,
                                       float* __restrict__ out) {
    __shared__ float tile[32][33];
    const int bh = blockIdx.x;            // 0..B*H-1
    const int b = bh / H_, h = bh % H_;
    const int c0 = blockIdx.y * 32;
    const int w0 = blockIdx.z * 32;
    const int tx = threadIdx.x;           // 0..31 (along W on read)
    const int ty = threadIdx.y;           // 0..7
#pragma unroll
    for (int r = 0; r < 4; ++r) {
        const int c = c0 + ty + r * 8;
        const int w = w0 + tx;
        if (c < C_ && w < W_)
            tile[ty + r * 8][tx] = in[((size_t)(b * C_ + c) * H_ + h) * W_ + w];
    }
    __syncthreads();
#pragma unroll
    for (int r = 0; r < 4; ++r) {
        const int w = w0 + ty + r * 8;
        const int c = c0 + tx;            // coalesced along C on write
        if (c < C_ && w < W_)
            out[((size_t)(b * H_ + h) * W_ + w) * C_ + c] = tile[tx][ty + r * 8];
    }
}

__global__ __launch_bounds__(256) void roi_align_pool_wmma(
    const float* __restrict__ feat,      // original BCHW (fallback path)
    const float* __restrict__ featT,     // BHWC (fast path)
    const float* __restrict__ rois,      // (N,5)
    float* __restrict__ out,             // (N,C,7,7)
    int useT) {
    __shared__ float Mlds[64 * MSTR];    // 49x64 pool matrix, zero-padded rows
    __shared__ float Vlds[64 * VSTR];    // 64 sample points x CHUNK channels
    __shared__ int   baseL[64];
    __shared__ float wgtL[64][4];

    const int n = blockIdx.x;
    const int t = threadIdx.x;

    // --- constant averaging matrix M[q][p]: 0.25 where the 2x2 window hits ---
    for (int idx = t; idx < 64 * 64; idx += 256) {
        const int q = idx >> 6, p = idx & 63;
        const int qh = q / 7, qw = q % 7;
        const int ph = p >> 3, pw = p & 7;
        float v = 0.0f;
        if (q < 49 && (ph == qh || ph == qh + 1) && (pw == qw || pw == qw + 1))
            v = 0.25f;
        Mlds[q * MSTR + p] = v;
    }

    // --- per-RoI sample coordinates / weights (64 points) ---
    if (t < 64) {
        const float* r = rois + (size_t)n * 5;
        const float S = 1.0f / 16.0f;
        const int   b  = (int)r[0];
        const float x1 = r[1] * S, y1 = r[2] * S;
        const float x2 = r[3] * S, y2 = r[4] * S;
        const float rw = fmaxf(x2 - x1 + 1.0f, 0.0f);
        const float rh = fmaxf(y2 - y1 + 1.0f, 0.0f);
        const float bh = rh / 7.0f, bw = rw / 7.0f;
        const int ph = t >> 3, pw = t & 7;
        const float hh = y1 + (float)ph * bh;
        const float ww = x1 + (float)pw * bw;
        const bool valid = (hh >= 0.0f) && (hh < (float)H_) &&
                           (ww >= 0.0f) && (ww < (float)W_);
        int hs = (int)floorf(hh); hs = min(max(hs, 0), H_ - 2);
        int ws = (int)floorf(ww); ws = min(max(ws, 0), W_ - 2);
        const float hr = hh - (float)hs, wr = ww - (float)ws;
        float w00 = (1.0f - hr) * (1.0f - wr);
        float w01 = (1.0f - hr) * wr;
        float w10 = hr * (1.0f - wr);
        float w11 = hr * wr;
        if (!valid) { w00 = w01 = w10 = w11 = 0.0f; }
        wgtL[t][0] = w00; wgtL[t][1] = w01; wgtL[t][2] = w10; wgtL[t][3] = w11;
        baseL[t] = useT ? ((b * H_ + hs) * W_ + ws) * C_
                        : (b * C_ * H_ * W_ + hs * W_ + ws);
    }
    __syncthreads();

    const int l    = t & 31;                 // lane in wave32
    const int wv   = t >> 5;                 // wave id 0..7
    const int cl   = wv * 16 + (l & 15);     // channel-in-chunk for WMMA phase
    const int cb   = t & (CHUNK - 1);        // channel-in-chunk for blend phase
    const int psel = t >> 7;                 // 0/1: which points this thread blends

    for (int c0 = 0; c0 < C_; c0 += CHUNK) {
        // ---- blend: gather 4 corners, weighted sum -> Vlds[p][c] ----
        for (int p = psel; p < 64; p += 2) {
            const int base = baseL[p];
            const float w00 = wgtL[p][0], w01 = wgtL[p][1];
            const float w10 = wgtL[p][2], w11 = wgtL[p][3];
            float f00, f01, f10, f11;
            if (useT) {
                const float* fp = featT + (size_t)base + c0 + cb;
                f00 = fp[0];
                f01 = fp[C_];
                f10 = fp[(size_t)W_ * C_];
                f11 = fp[(size_t)W_ * C_ + C_];
            } else {
                const float* fp = feat + (size_t)base + (size_t)(c0 + cb) * (H_ * W_);
                f00 = fp[0];
                f01 = fp[1];
                f10 = fp[W_];
                f11 = fp[W_ + 1];
            }
            Vlds[p * VSTR + cb] = w00 * f00 + w01 * f01 + w10 * f10 + w11 * f11;
        }
        __syncthreads();

        // ---- pool: O(64x16) = M(64x64) x V(64x16ch) via chained f32 WMMA ----
        for (int mt = 0; mt < 4; ++mt) {
            v8f acc = {};
            for (int kk = 0; kk < 16; ++kk) {
                const int kb = 4 * kk + ((l >> 4) << 1);
                v2f a, bm;
                a.x  = Mlds[(mt * 16 + (l & 15)) * MSTR + kb];
                a.y  = Mlds[(mt * 16 + (l & 15)) * MSTR + kb + 1];
                bm.x = Vlds[(kb + 0) * VSTR + cl];
                bm.y = Vlds[(kb + 1) * VSTR + cl];
                acc = __builtin_amdgcn_wmma_f32_16x16x4_f32(
                    /*neg_a=*/false, a, /*neg_b=*/false, bm,
                    /*c_mod=*/(short)0, acc, /*reuse_a=*/false, /*reuse_b=*/false);
            }
            const int c = c0 + cl;
            const size_t ob = (size_t)n * C_ * 49 + (size_t)c * 49;
            const int rofs = (l >= 16) ? 8 : 0;
#pragma unroll
            for (int m = 0; m < 8; ++m) {
                const int q = mt * 16 + m + rofs;   // output index qh*7+qw
                if (q < 49) out[ob + q] = acc[m];
            }
        }
        __syncthreads();
    }
}

extern "C" void kernel_launch(void* const* d_in, const int* in_sizes, int n_in,
                              void* d_out, int out_size, void* d_ws, size_t ws_size,
                              hipStream_t stream) {
    const float* feat = (const float*)d_in[0];
    const float* rois = (const float*)d_in[1];
    float* out = (float*)d_out;
    const int N = in_sizes[1] / 5;

    const size_t needT = (size_t)B_ * C_ * H_ * W_ * sizeof(float);
    const int useT = (d_ws != nullptr && ws_size >= needT) ? 1 : 0;
    const float* featT = useT ? (const float*)d_ws : feat;

    if (useT) {
        dim3 tb(32, 8, 1);
        dim3 tg(B_ * H_, (C_ + 31) / 32, (W_ + 31) / 32);
        transpose_bchw_to_bhwc<<<tg, tb, 0, stream>>>(feat, (float*)d_ws);
    }
    roi_align_pool_wmma<<<dim3(N, 1, 1), dim3(256, 1, 1), 0, stream>>>(
        feat, featT, rois, out, useT);
}